// GCN_40896678592994
// MI455X (gfx1250) — compile-verified
//
#include <hip/hip_runtime.h>
#include <hip/hip_bf16.h>

typedef float v2f __attribute__((ext_vector_type(2)));
typedef float v8f __attribute__((ext_vector_type(8)));

// ---------------------------------------------------------------------------
// Degree / normalization
// ---------------------------------------------------------------------------
__global__ void gcn_deg_init(float* __restrict__ deg, int n) {
  int i = blockIdx.x * blockDim.x + threadIdx.x;
  if (i < n) deg[i] = 1.0f;  // self-loop contributes 1 to every node's degree
}

__global__ void gcn_deg_count(const int* __restrict__ dst, float* __restrict__ deg, int e) {
  int i = blockIdx.x * blockDim.x + threadIdx.x;
  if (i < e) atomicAdd(&deg[dst[i]], 1.0f);
}

__global__ void gcn_dinv(float* __restrict__ deg, int n) {
  int i = blockIdx.x * blockDim.x + threadIdx.x;
  if (i < n) deg[i] = rsqrtf(deg[i]);  // deg >= 1 always (self-loop)
}

// ---------------------------------------------------------------------------
// Layer-1 matmul: hlin[N,4] = x[N,128] @ W1[128,4] via V_WMMA_F32_16X16X4_F32.
// One wave32 computes a 16-row output tile; K fully unrolled: 32 WMMA steps.
// B (4->16 zero-padded) is staged in LDS once per block so the inner loop has
// ZERO divergent branches: per step it is one global_load_b64 (A, the real
// bandwidth carrier) + two ds_load_b32 (B) + one v_wmma.
// Fragment layouts per CDNA5 ISA 7.12.2 (32-bit A 16x4, 32-bit B 4x16, C 16x16).
// ---------------------------------------------------------------------------
__global__ void gcn_matmul1_wmma(const float* __restrict__ x,
                                 const float* __restrict__ W1,  // [128,4] row-major
                                 float* __restrict__ hlin,      // [N,4]
                                 int nNodes) {
  __shared__ float Bpad[128 * 16];  // Bpad[k][n] = (n<4) ? W1[k][n] : 0

  // Cooperative zero-padded fill: 2048 floats / 256 threads = 8 each.
  for (int idx = threadIdx.x; idx < 128 * 16; idx += blockDim.x) {
    const int k = idx >> 4;
    const int n = idx & 15;
    Bpad[idx] = (n < 4) ? W1[k * 4 + n] : 0.0f;
  }
  __syncthreads();

  const int wave  = (blockIdx.x * blockDim.x + threadIdx.x) >> 5;  // wave32
  const int lane  = threadIdx.x & 31;
  const int row0  = wave * 16;
  if (row0 >= nNodes) return;  // wave-uniform: EXEC stays all-ones for live waves

  const int m     = lane & 15;   // A row / B col / D col within tile
  const int khalf = lane >> 4;   // 0: K pair {0,1}; 1: K pair {2,3}

  // Clamp row for safety (N=100000 is a multiple of 16, so never triggers).
  int arow = row0 + m;
  if (arow > nNodes - 1) arow = nNodes - 1;
  const float* xrow = x + (size_t)arow * 128;
  const float* brow = Bpad + m + (size_t)khalf * 2 * 16;  // row kb = k + 2*khalf

  v8f c = {};  // fp32 accumulator, 8 VGPRs
#pragma unroll
  for (int k = 0; k < 128; k += 4) {
    const int kb = k + khalf * 2;
    v2f a = *(const v2f*)(xrow + kb);  // contiguous 8B load per lane
    v2f b;
    b.x = brow[(size_t)k * 16];        // Bpad[kb][m]
    b.y = brow[(size_t)k * 16 + 16];   // Bpad[kb+1][m]
    c = __builtin_amdgcn_wmma_f32_16x16x4_f32(
        /*neg_a=*/false, a, /*neg_b=*/false, b,
        /*c_mod=*/(short)0, c, /*reuse_a=*/false, /*reuse_b=*/false);
  }

  // D layout: VGPR r, lane L -> (M = r + 8*(L>>4), N = L&15); only N<4 valid.
  if (m < 4) {
    const int mbase = row0 + 8 * khalf;
#pragma unroll
    for (int r = 0; r < 8; ++r) {
      const int row = mbase + r;
      if (row < nNodes) hlin[(size_t)row * 4 + m] = c[r];
    }
  }
}

// ---------------------------------------------------------------------------
// Tiny per-node matmul (kin,kout <= 4): hout[N,kout] = hin[N,kin] @ W[kin,kout]
// ---------------------------------------------------------------------------
__global__ void gcn_matmul_small(const float* __restrict__ hin,
                                 const float* __restrict__ W,
                                 float* __restrict__ hout,
                                 int n, int kin, int kout) {
  int i = blockIdx.x * blockDim.x + threadIdx.x;
  if (i >= n) return;
  float in[4];
  for (int c = 0; c < kin; ++c) in[c] = hin[(size_t)i * kin + c];
  for (int j = 0; j < kout; ++j) {
    float acc = 0.0f;
    for (int c = 0; c < kin; ++c) acc = fmaf(in[c], W[c * kout + j], acc);
    hout[(size_t)i * kout + j] = acc;
  }
}

// ---------------------------------------------------------------------------
// agg[i] = bias + dinv[i]^2 * hlin[i]   (self-loop edge fused with bias init)
// ---------------------------------------------------------------------------
__global__ void gcn_init_agg(const float* __restrict__ hlin,
                             const float* __restrict__ dinv,
                             const float* __restrict__ bias,
                             float* __restrict__ agg,
                             int n, int width) {
  int i = blockIdx.x * blockDim.x + threadIdx.x;
  if (i >= n) return;
  const float w = dinv[i] * dinv[i];
  for (int c = 0; c < width; ++c)
    agg[(size_t)i * width + c] = fmaf(w, hlin[(size_t)i * width + c], bias[c]);
}

// ---------------------------------------------------------------------------
// Edge scatter: agg[dst] += dinv[src]*dinv[dst] * hlin[src]  (width 4 and 2)
// ---------------------------------------------------------------------------
__global__ void gcn_scatter4(const float* __restrict__ hlin,
                             const int* __restrict__ src,
                             const int* __restrict__ dst,
                             const float* __restrict__ dinv,
                             float* __restrict__ agg, int e) {
  int i = blockIdx.x * blockDim.x + threadIdx.x;
  if (i >= e) return;
  const int s = src[i], d = dst[i];
  const float w = dinv[s] * dinv[d];
  const float4 h = *(const float4*)(hlin + (size_t)s * 4);
  float* ap = agg + (size_t)d * 4;
  atomicAdd(ap + 0, w * h.x);
  atomicAdd(ap + 1, w * h.y);
  atomicAdd(ap + 2, w * h.z);
  atomicAdd(ap + 3, w * h.w);
}

__global__ void gcn_scatter2(const float* __restrict__ hlin,
                             const int* __restrict__ src,
                             const int* __restrict__ dst,
                             const float* __restrict__ dinv,
                             float* __restrict__ agg, int e) {
  int i = blockIdx.x * blockDim.x + threadIdx.x;
  if (i >= e) return;
  const int s = src[i], d = dst[i];
  const float w = dinv[s] * dinv[d];
  const float2 h = *(const float2*)(hlin + (size_t)s * 2);
  float* ap = agg + (size_t)d * 2;
  atomicAdd(ap + 0, w * h.x);
  atomicAdd(ap + 1, w * h.y);
}

// ---------------------------------------------------------------------------
// Elementwise tanh; optionally mirrors result into a second buffer (layer-3 h)
// ---------------------------------------------------------------------------
__global__ void gcn_tanh(const float* __restrict__ in, float* __restrict__ out,
                         float* __restrict__ out2, int total) {
  int i = blockIdx.x * blockDim.x + threadIdx.x;
  if (i >= total) return;
  const float v = tanhf(in[i]);
  out[i] = v;
  if (out2) out2[i] = v;
}

// ---------------------------------------------------------------------------
// Classifier: out[N,10] = h[N,2] @ Wc[2,10] + bc
// ---------------------------------------------------------------------------
__global__ void gcn_classifier(const float* __restrict__ h,
                               const float* __restrict__ Wc,
                               const float* __restrict__ bc,
                               float* __restrict__ out, int n) {
  int i = blockIdx.x * blockDim.x + threadIdx.x;
  if (i >= n) return;
  const float h0 = h[(size_t)i * 2 + 0];
  const float h1 = h[(size_t)i * 2 + 1];
  float* op = out + (size_t)i * 10;
#pragma unroll
  for (int j = 0; j < 10; ++j)
    op[j] = fmaf(h0, Wc[j], fmaf(h1, Wc[10 + j], bc[j]));
}

// ---------------------------------------------------------------------------
extern "C" void kernel_launch(void* const* d_in, const int* in_sizes, int n_in,
                              void* d_out, int out_size, void* d_ws, size_t ws_size,
                              hipStream_t stream) {
  const float* x    = (const float*)d_in[0];
  const int*   edge = (const int*)d_in[1];   // [2, E] int32
  const float* W1   = (const float*)d_in[2];
  const float* b1   = (const float*)d_in[3];
  const float* W2   = (const float*)d_in[4];
  const float* b2   = (const float*)d_in[5];
  const float* W3   = (const float*)d_in[6];
  const float* b3   = (const float*)d_in[7];
  const float* Wc   = (const float*)d_in[8];
  const float* bc   = (const float*)d_in[9];

  const int N = in_sizes[0] / 128;
  const int E = in_sizes[1] / 2;
  const int* srcp = edge;
  const int* dstp = edge + E;

  // Workspace layout (floats): dinv[N] | hlin[4N] | hact[4N] | agg[4N]
  float* ws   = (float*)d_ws;
  float* dinv = ws;
  float* hlin = ws + (size_t)N;
  float* hact = ws + (size_t)5 * N;
  float* agg  = ws + (size_t)9 * N;

  float* out_logits = (float*)d_out;                // [N,10]
  float* out_h      = out_logits + (size_t)N * 10;  // [N,2]

  const int B  = 256;
  const int gN = (N + B - 1) / B;
  const int gE = (E + B - 1) / B;

  // --- normalization ---
  gcn_deg_init <<<gN, B, 0, stream>>>(dinv, N);
  gcn_deg_count<<<gE, B, 0, stream>>>(dstp, dinv, E);
  gcn_dinv     <<<gN, B, 0, stream>>>(dinv, N);

  // --- layer 1: WMMA matmul (128 -> 4), conv, tanh ---
  const int tiles   = (N + 15) / 16;
  const int wmmaBlk = (tiles + 7) / 8;  // 8 wave32 per 256-thread block
  gcn_matmul1_wmma<<<wmmaBlk, B, 0, stream>>>(x, W1, hlin, N);
  gcn_init_agg<<<gN, B, 0, stream>>>(hlin, dinv, b1, agg, N, 4);
  gcn_scatter4<<<gE, B, 0, stream>>>(hlin, srcp, dstp, dinv, agg, E);
  gcn_tanh<<<(4 * N + B - 1) / B, B, 0, stream>>>(agg, hact, nullptr, 4 * N);

  // --- layer 2: 4 -> 4 ---
  gcn_matmul_small<<<gN, B, 0, stream>>>(hact, W2, hlin, N, 4, 4);
  gcn_init_agg<<<gN, B, 0, stream>>>(hlin, dinv, b2, agg, N, 4);
  gcn_scatter4<<<gE, B, 0, stream>>>(hlin, srcp, dstp, dinv, agg, E);
  gcn_tanh<<<(4 * N + B - 1) / B, B, 0, stream>>>(agg, hact, nullptr, 4 * N);

  // --- layer 3: 4 -> 2 (tanh also writes h output) ---
  gcn_matmul_small<<<gN, B, 0, stream>>>(hact, W3, hlin, N, 4, 2);
  gcn_init_agg<<<gN, B, 0, stream>>>(hlin, dinv, b3, agg, N, 2);
  gcn_scatter2<<<gE, B, 0, stream>>>(hlin, srcp, dstp, dinv, agg, E);
  gcn_tanh<<<(2 * N + B - 1) / B, B, 0, stream>>>(agg, hact, out_h, 2 * N);

  // --- classifier: 2 -> 10 ---
  gcn_classifier<<<gN, B, 0, stream>>>(hact, Wc, bc, out_logits, N);
}